// TwoLayerGCN_89790586290635
// MI455X (gfx1250) — compile-verified
//
#include <hip/hip_runtime.h>
#include <hip/hip_bf16.h>
#include <math.h>

// ---------------------------------------------------------------------------
// TwoLayerGCN on MI455X (gfx1250, wave32).
//   layer: h = x @ W ; out[i] = sum_{e: dst=i} h[src]*dinv[src]*dinv[dst] + b
//   (self-loops included; dinv = rsqrt(degree), degree >= 1)
// GEMMs use V_WMMA_F32_16X16X4_F32 (exact fp32 matrix path).
// Aggregation uses L2-resident f32 atomics (footprint << 192MB L2).
// ---------------------------------------------------------------------------

typedef float v2f __attribute__((ext_vector_type(2)));
typedef float v8f __attribute__((ext_vector_type(8)));

#define HID 64  // both layer output dims are 64

// ---- degree / dinv ---------------------------------------------------------
__global__ void k_deg_init(float* __restrict__ deg, int n) {
  int i = blockIdx.x * blockDim.x + threadIdx.x;
  if (i < n) deg[i] = 1.0f;  // self-loop contribution
}

__global__ void k_deg_scatter(const long long* __restrict__ dst,
                              float* __restrict__ deg, int ne) {
  int i = blockIdx.x * blockDim.x + threadIdx.x;
  if (i < ne) unsafeAtomicAdd(&deg[(int)dst[i]], 1.0f);
}

__global__ void k_rsqrt_inplace(float* __restrict__ d, int n) {
  int i = blockIdx.x * blockDim.x + threadIdx.x;
  if (i < n) d[i] = rsqrtf(d[i]);
}

// ---- WMMA GEMM: H[M,64] = act(A[M,K]) @ W[K,64] ---------------------------
// One wave computes a 16x64 tile. A-frag 16x4 f32 (v2f/lane), B-frag 4x16
// (v2f/lane, lanes index N), C/D 16x16 f32 (v8f/lane). M must be 16*ntiles.
template <int K, bool RELU_IN>
__global__ void k_gemm_wmma(const float* __restrict__ A,
                            const float* __restrict__ W,
                            float* __restrict__ H, int ntiles) {
  const int lane = threadIdx.x & 31;
  const int wv   = threadIdx.x >> 5;
  const int mt   = blockIdx.x * 8 + wv;     // wave-uniform -> EXEC all-1s inside
  if (mt >= ntiles) return;

  const int half = lane >> 4;               // 0: K pair {0,1}, 1: {2,3}
  const int l16  = lane & 15;
  const int arow = mt * 16 + l16;

  v8f acc[4] = {};                          // 4 N-tiles of 16 cols each

  for (int kk = 0; kk < K; kk += 4) {
    const int kb = kk + 2 * half;
    // A fragment: row = arow, cols kb,kb+1 (contiguous -> b64 load)
    const float* ap = A + (size_t)arow * K + kb;
    float a0 = ap[0], a1 = ap[1];
    if (RELU_IN) { a0 = fmaxf(a0, 0.0f); a1 = fmaxf(a1, 0.0f); }
    v2f a; a.x = a0; a.y = a1;
#pragma unroll
    for (int nt = 0; nt < 4; ++nt) {
      // B fragment: rows kb,kb+1 of W, col = nt*16 + l16
      v2f b;
      b.x = W[(size_t)kb       * HID + nt * 16 + l16];
      b.y = W[(size_t)(kb + 1) * HID + nt * 16 + l16];
      // (neg_a, A, neg_b, B, c_mod, C, reuse_a, reuse_b)
      acc[nt] = __builtin_amdgcn_wmma_f32_16x16x4_f32(
          false, a, false, b, (short)0, acc[nt], false, false);
    }
  }

  // D layout: lane<16 -> M = r (vgpr index), lane>=16 -> M = 8 + r; N = l16
  const int rbase = mt * 16 + 8 * half;
#pragma unroll
  for (int r = 0; r < 8; ++r) {
    float* op = H + (size_t)(rbase + r) * HID + l16;
    op[0]  = acc[0][r];
    op[16] = acc[1][r];
    op[32] = acc[2][r];
    op[48] = acc[3][r];
  }
}

// ---- aggregation -----------------------------------------------------------
// init: out = self-loop term + bias (bias added exactly once here)
__global__ void k_agg_init(const float* __restrict__ h,
                           const float* __restrict__ dinv,
                           const float* __restrict__ bias,
                           float* __restrict__ out, int n) {
  int i = blockIdx.x * blockDim.x + threadIdx.x;
  if (i >= n * HID) return;
  int node = i >> 6;
  int d    = i & 63;
  float dv = dinv[node];
  out[i] = h[i] * dv * dv + bias[d];
}

// scatter: 16 threads per edge, float4 gather + 4 hw f32 atomics (L2-resident)
__global__ void k_agg_scatter(const float* __restrict__ h,
                              const long long* __restrict__ src,
                              const long long* __restrict__ dst,
                              const float* __restrict__ dinv,
                              float* __restrict__ out, int ne) {
  long long tid = (long long)blockIdx.x * blockDim.x + threadIdx.x;
  if (tid >= (long long)ne * 16) return;
  int e  = (int)(tid >> 4);
  int d4 = (int)(tid & 15) << 2;
  int s = (int)src[e];
  int t = (int)dst[e];
  float norm = dinv[s] * dinv[t];
  const float4 v = *reinterpret_cast<const float4*>(h + (size_t)s * HID + d4);
  float* o = out + (size_t)t * HID + d4;
  unsafeAtomicAdd(o + 0, v.x * norm);
  unsafeAtomicAdd(o + 1, v.y * norm);
  unsafeAtomicAdd(o + 2, v.z * norm);
  unsafeAtomicAdd(o + 3, v.w * norm);
}

// ---------------------------------------------------------------------------
extern "C" void kernel_launch(void* const* d_in, const int* in_sizes, int n_in,
                              void* d_out, int out_size, void* d_ws,
                              size_t ws_size, hipStream_t stream) {
  const float*     x   = (const float*)d_in[0];
  const long long* ei  = (const long long*)d_in[1];  // int64 per reference
  const float*     W1  = (const float*)d_in[2];
  const float*     b1  = (const float*)d_in[3];
  const float*     W2  = (const float*)d_in[4];
  const float*     b2  = (const float*)d_in[5];
  float*           out = (float*)d_out;

  const int nNodes = in_sizes[0] / 128;  // 50000
  const int nEdges = in_sizes[1] / 2;    // 800000
  const long long* srcp = ei;            // edge_index[0]
  const long long* dstp = ei + nEdges;   // edge_index[1]

  // workspace layout (25.8 MB): dinv | h (reused for both layers) | ag1
  char*  ws   = (char*)d_ws;
  size_t off  = 0;
  float* dinv = (float*)(ws + off);
  off += (((size_t)nNodes * 4) + 255) & ~(size_t)255;
  float* h1 = (float*)(ws + off);
  off += (((size_t)nNodes * HID * 4) + 255) & ~(size_t)255;
  float* ag1 = (float*)(ws + off);
  float* h2  = h1;  // h1 is dead after layer-1 aggregation

  const int B = 256;
  const int ntiles = nNodes / 16;  // 50000 = 16 * 3125, exact
  const int gemmBlocks = (ntiles + 7) / 8;
  const int elemBlocks = (nNodes * HID + B - 1) / B;
  const unsigned scatBlocks =
      (unsigned)(((long long)nEdges * 16 + B - 1) / B);

  // dinv = rsqrt(1 + in-degree)
  k_deg_init<<<(nNodes + B - 1) / B, B, 0, stream>>>(dinv, nNodes);
  k_deg_scatter<<<(nEdges + B - 1) / B, B, 0, stream>>>(dstp, dinv, nEdges);
  k_rsqrt_inplace<<<(nNodes + B - 1) / B, B, 0, stream>>>(dinv, nNodes);

  // layer 1: h1 = x @ W1 ; ag1 = aggregate(h1) + b1
  k_gemm_wmma<128, false><<<gemmBlocks, B, 0, stream>>>(x, W1, h1, ntiles);
  k_agg_init<<<elemBlocks, B, 0, stream>>>(h1, dinv, b1, ag1, nNodes);
  k_agg_scatter<<<scatBlocks, B, 0, stream>>>(h1, srcp, dstp, dinv, ag1,
                                              nEdges);

  // layer 2: h2 = relu(ag1) @ W2 (ReLU fused into A load) ; out = agg + b2
  k_gemm_wmma<64, true><<<gemmBlocks, B, 0, stream>>>(ag1, W2, h2, ntiles);
  k_agg_init<<<elemBlocks, B, 0, stream>>>(h2, dinv, b2, out, nNodes);
  k_agg_scatter<<<scatBlocks, B, 0, stream>>>(h2, srcp, dstp, dinv, out,
                                              nEdges);
}